// Conv2D_2164663517649
// MI455X (gfx1250) — compile-verified
//
#include <hip/hip_runtime.h>

typedef float v2f __attribute__((ext_vector_type(2)));
typedef float v8f __attribute__((ext_vector_type(8)));

#define IH 4096
#define IW 4096
#define KH 15
#define KW 15
#define OH (IH - KH + 1) /* 4082 */
#define OW (IW - KW + 1) /* 4082 */

#define NWAVES 8
#define TJ 4                              /* horizontal 16-col tiles per wave */
#define BLOCK_ROWS (NWAVES * 16)          /* 128 output rows per block  */
#define BLOCK_COLS (TJ * 16)              /* 64 output cols per block   */
#define REG_ROWS (BLOCK_ROWS + KH - 1)    /* 142 X rows staged          */
#define REG_COLS (BLOCK_COLS + 16)        /* 80: last tile reads +31    */
#define XSTRIDE 82                        /* even (b64 align), conflict-free */
#define WSTRIDE 48                        /* 16 zero | 15 taps | 17 zero */

__global__ __launch_bounds__(256) void conv15_wmma_f32(
    const float* __restrict__ X, const float* __restrict__ Wgt,
    const float* __restrict__ bias, float* __restrict__ out) {
  __shared__ float sX[REG_ROWS * XSTRIDE];
  __shared__ float sW[KH * WSTRIDE];

  const int tid = threadIdx.x;
  const int c0 = blockIdx.x * BLOCK_COLS;  // output col base of this block
  const int r0 = blockIdx.y * BLOCK_ROWS;  // output row base of this block

  // Zero-padded Toeplitz rows of the kernel: sW[p][16+d] = W[p][d], d in [0,15)
  for (int idx = tid; idx < KH * WSTRIDE; idx += 256) {
    const int p = idx / WSTRIDE;
    const int d = (idx % WSTRIDE) - 16;
    sW[idx] = (d >= 0 && d < KW) ? Wgt[p * KW + d] : 0.0f;
  }
  // Stage the 142x80 X slab. Clamped loads at bottom/right edges: clamped
  // values either multiply the zero band of sW or feed tiles whose outputs
  // are never stored. All 80 cols are staged with real data (never garbage),
  // so the zero band can't see NaNs.
  for (int idx = tid; idx < REG_ROWS * REG_COLS; idx += 256) {
    const int r = idx / REG_COLS;
    const int c = idx % REG_COLS;
    int gr = r0 + r; if (gr > IH - 1) gr = IH - 1;
    int gc = c0 + c; if (gc > IW - 1) gc = IW - 1;
    sX[r * XSTRIDE + c] = X[(size_t)gr * IW + gc];
  }
  __syncthreads();

  const int lane = tid & 31;   // wave32
  const int wave = tid >> 5;
  const int m  = lane & 15;    // A: row M / B: col N / C: col N
  const int hi = lane >> 4;    // lane half selects K {+0,+1} vs {+2,+3}

  v8f acc0 = {}, acc1 = {}, acc2 = {}, acc3 = {};

  for (int p = 0; p < KH; ++p) {
    // A rows for this lane: X[i0 + m + p, j0 + 16*t + k]
    const float* xr = &sX[(wave * 16 + m + p) * XSTRIDE];
    // B band for this lane: W[p, k - n], n == m ; index 16 + k - n in [1,47]
    const float* wr = &sW[p * WSTRIDE + 16 - m];
#pragma unroll
    for (int kc = 0; kc < 8; ++kc) {
      const int ck = 4 * kc + 2 * hi;
      v2f B; B.x = wr[ck]; B.y = wr[ck + 1];   // shared by all 4 tiles
      v2f A0; A0.x = xr[ck];      A0.y = xr[ck + 1];
      v2f A1; A1.x = xr[16 + ck]; A1.y = xr[16 + ck + 1];
      v2f A2; A2.x = xr[32 + ck]; A2.y = xr[32 + ck + 1];
      v2f A3; A3.x = xr[48 + ck]; A3.y = xr[48 + ck + 1];
      acc0 = __builtin_amdgcn_wmma_f32_16x16x4_f32(
          false, A0, false, B, (short)0, acc0, false, false);
      acc1 = __builtin_amdgcn_wmma_f32_16x16x4_f32(
          false, A1, false, B, (short)0, acc1, false, false);
      acc2 = __builtin_amdgcn_wmma_f32_16x16x4_f32(
          false, A2, false, B, (short)0, acc2, false, false);
      acc3 = __builtin_amdgcn_wmma_f32_16x16x4_f32(
          false, A3, false, B, (short)0, acc3, false, false);
    }
  }

  // C/D layout: VGPR v -> row (v + 8*hi), col n(=m)
  const float bs = bias[0];
  const int ibase = r0 + wave * 16 + 8 * hi;
#pragma unroll
  for (int t = 0; t < TJ; ++t) {
    const int j = c0 + 16 * t + m;
    if (j < OW) {
      const v8f acc = (t == 0) ? acc0 : (t == 1) ? acc1 : (t == 2) ? acc2 : acc3;
#pragma unroll
      for (int v = 0; v < 8; ++v) {
        const int i = ibase + v;
        if (i < OH) out[(size_t)i * OW + j] = acc[v] + bs;
      }
    }
  }
}

extern "C" void kernel_launch(void* const* d_in, const int* in_sizes, int n_in,
                              void* d_out, int out_size, void* d_ws, size_t ws_size,
                              hipStream_t stream) {
  (void)in_sizes; (void)n_in; (void)out_size; (void)d_ws; (void)ws_size;
  const float* X    = (const float*)d_in[0];
  const float* Wgt  = (const float*)d_in[1];
  const float* bias = (const float*)d_in[2];
  float* out        = (float*)d_out;

  dim3 block(256);
  dim3 grid((OW + BLOCK_COLS - 1) / BLOCK_COLS,    // 64
            (OH + BLOCK_ROWS - 1) / BLOCK_ROWS);   // 32
  conv15_wmma_f32<<<grid, block, 0, stream>>>(X, Wgt, bias, out);
}